// SimpleGCN_6640019440134
// MI455X (gfx1250) — compile-verified
//
#include <hip/hip_runtime.h>
#include <hip/hip_bf16.h>

// CDNA5 / gfx1250 GCN pipeline.
// Memory-bound problem (≈2 GB traffic vs ≈2 GFLOP) -> keep fp32 end-to-end and
// use V_WMMA_F32_16X16X4_F32 for the 64x64 GEMMs (full precision, matrix pipe).

typedef __attribute__((ext_vector_type(2))) float v2f;
typedef __attribute__((ext_vector_type(8))) float v8f;

__device__ __forceinline__ void atomic_add_f32(float* p, float v) {
  // lowers to global_atomic_add_f32 (agent scope, relaxed)
  __hip_atomic_fetch_add(p, v, __ATOMIC_RELAXED, __HIP_MEMORY_SCOPE_AGENT);
}

// ---------------------------------------------------------------- utilities
__global__ void k_zero(float* __restrict__ p, int n) {
  int i = blockIdx.x * 256 + threadIdx.x;
  if (i < n) p[i] = 0.0f;
}

__global__ void k_count_deg(const int* __restrict__ dst, int* __restrict__ deg, int e) {
  int i = blockIdx.x * 256 + threadIdx.x;
  if (i < e) atomicAdd(&deg[dst[i]], 1);
}

__global__ void k_dis(const int* __restrict__ deg, float* __restrict__ dis, int n) {
  int i = blockIdx.x * 256 + threadIdx.x;
  if (i < n) dis[i] = rsqrtf((float)(deg[i] + 1));  // +1 self loop; always > 0
}

// ------------------------------------------------- layer 1: X[N,7] @ W1[7,64]
__global__ __launch_bounds__(256) void k_linear_in(const float* __restrict__ x,
                                                   const float* __restrict__ W1,
                                                   float* __restrict__ Y, int n) {
  long long idx = (long long)blockIdx.x * 256 + threadIdx.x;
  if (idx >= (long long)n * 64) return;
  int row = (int)(idx >> 6), c = (int)(idx & 63);
  const float* xr = x + (size_t)row * 7;
  float acc = 0.0f;
#pragma unroll
  for (int k = 0; k < 7; ++k) acc += xr[k] * W1[k * 64 + c];
  Y[idx] = acc;
}

// --------------------------------- layers 2/3: H[N,64] @ W[64,64] via WMMA f32
// Block = 256 threads = 8 waves; block owns 128 rows (16 rows per wave).
// LDS: X tile 128x64 (pitch 65) + W transposed 64x64 (pitch 65)  ~= 49.9 KB.
// Per wave: 4 accumulator tiles (16x16 each) over the 64-col output,
// K=64 consumed as 16 steps of V_WMMA_F32_16X16X4_F32 per tile.
__global__ __launch_bounds__(256) void k_linear_wmma(const float* __restrict__ X,
                                                     const float* __restrict__ W,
                                                     float* __restrict__ Y, int n) {
  __shared__ float ldsX[128 * 65];
  __shared__ float ldsWt[64 * 65];
  const int tid = threadIdx.x;
  const int rowBase = blockIdx.x * 128;

  // W[k][c] -> ldsWt[c][k] (transposed so B fragments are contiguous 8B loads)
  for (int i = tid; i < 64 * 64; i += 256) {
    int k = i >> 6, c = i & 63;
    ldsWt[c * 65 + k] = W[i];
  }
  // X tile, coalesced, zero-padded past N
  for (int i = tid; i < 128 * 64; i += 256) {
    int r = i >> 6, c = i & 63;
    int gr = rowBase + r;
    ldsX[r * 65 + c] = (gr < n) ? X[(size_t)gr * 64 + c] : 0.0f;
  }
  __syncthreads();

  const int wave = tid >> 5;
  const int lane = tid & 31;
  const int m  = lane & 15;          // M (A) / N (B) index within 16
  const int kh = (lane >> 4) << 1;   // lanes 0-15: K=0,1  lanes 16-31: K=2,3

  const float* xr = &ldsX[(wave * 16 + m) * 65];
  const float* w0 = &ldsWt[(0  + m) * 65];
  const float* w1 = &ldsWt[(16 + m) * 65];
  const float* w2 = &ldsWt[(32 + m) * 65];
  const float* w3 = &ldsWt[(48 + m) * 65];

  v8f c0 = {}, c1 = {}, c2 = {}, c3 = {};
#pragma unroll
  for (int kb = 0; kb < 16; ++kb) {
    const int k = kb * 4 + kh;
    v2f a;  a.x  = xr[k]; a.y  = xr[k + 1];
    v2f b0; b0.x = w0[k]; b0.y = w0[k + 1];
    v2f b1; b1.x = w1[k]; b1.y = w1[k + 1];
    v2f b2; b2.x = w2[k]; b2.y = w2[k + 1];
    v2f b3; b3.x = w3[k]; b3.y = w3[k + 1];
    // (neg_a, A, neg_b, B, c_mod, C, reuse_a, reuse_b)
    c0 = __builtin_amdgcn_wmma_f32_16x16x4_f32(false, a, false, b0, (short)0, c0, false, false);
    c1 = __builtin_amdgcn_wmma_f32_16x16x4_f32(false, a, false, b1, (short)0, c1, false, false);
    c2 = __builtin_amdgcn_wmma_f32_16x16x4_f32(false, a, false, b2, (short)0, c2, false, false);
    c3 = __builtin_amdgcn_wmma_f32_16x16x4_f32(false, a, false, b3, (short)0, c3, false, false);
  }

  // C/D layout: VGPR j, lanes 0-15 -> M=j, lanes 16-31 -> M=j+8; N = lane&15
  const int r0 = rowBase + wave * 16 + ((lane >> 4) << 3);
#pragma unroll
  for (int j = 0; j < 8; ++j) {
    int r = r0 + j;
    if (r < n) {
      float* yr = Y + (size_t)r * 64;
      yr[m]      = c0[j];
      yr[16 + m] = c1[j];
      yr[32 + m] = c2[j];
      yr[48 + m] = c3[j];
    }
  }
}

// --------------------------- self-loop init: agg[n,c] = XW[n,c] * dis[n]^2
__global__ void k_init_agg(const float* __restrict__ xw, const float* __restrict__ dis,
                           float* __restrict__ agg, int n) {
  long long idx = (long long)blockIdx.x * 256 + threadIdx.x;
  if (idx >= (long long)n * 64) return;
  float d = dis[(int)(idx >> 6)];
  agg[idx] = xw[idx] * d * d;
}

// --------------------------- edge scatter: agg[dst] += XW[src]*dis[s]*dis[d]
// 16 lanes per edge, float4 each -> one contiguous 256B row read per edge.
__global__ __launch_bounds__(256) void k_scatter(const float* __restrict__ hsrc,
                                                 const int* __restrict__ src,
                                                 const int* __restrict__ dst,
                                                 const float* __restrict__ dis,
                                                 float* __restrict__ agg, int e) {
  long long gid = (long long)blockIdx.x * 256 + threadIdx.x;
  if (gid >= (long long)e * 16) return;
  int ei = (int)(gid >> 4);
  int q  = (int)(gid & 15);
  int s = src[ei], d = dst[ei];
  float w = dis[s] * dis[d];
  const float4 v = *(const float4*)(hsrc + (size_t)s * 64 + q * 4);
  float* o = agg + (size_t)d * 64 + q * 4;
  atomic_add_f32(o + 0, v.x * w);
  atomic_add_f32(o + 1, v.y * w);
  atomic_add_f32(o + 2, v.z * w);
  atomic_add_f32(o + 3, v.w * w);
}

// ----------------------------------- BN stats: per-column sum / sum-of-squares
__global__ __launch_bounds__(256) void k_stats(const float* __restrict__ h,
                                               const float* __restrict__ bias,
                                               float* __restrict__ stats, int n) {
  __shared__ float s_sum[256], s_sq[256];
  int tid = threadIdx.x;
  int c = tid & 63, rg = tid >> 6;
  int base = blockIdx.x * 512;
  int mend = min(base + 512, n);
  float b = bias[c];
  float s = 0.0f, q = 0.0f;
  for (int r = base + rg; r < mend; r += 4) {
    float v = h[(size_t)r * 64 + c] + b;
    s += v; q += v * v;
  }
  s_sum[tid] = s; s_sq[tid] = q;
  __syncthreads();
  if (tid < 64) {
    float ts = s_sum[tid] + s_sum[tid + 64] + s_sum[tid + 128] + s_sum[tid + 192];
    float tq = s_sq[tid] + s_sq[tid + 64] + s_sq[tid + 128] + s_sq[tid + 192];
    atomic_add_f32(&stats[tid], ts);
    atomic_add_f32(&stats[64 + tid], tq);
  }
}

__global__ void k_bn_relu(float* __restrict__ h, const float* __restrict__ bias,
                          const float* __restrict__ stats, const float* __restrict__ gamma,
                          const float* __restrict__ beta, int n) {
  long long idx = (long long)blockIdx.x * 256 + threadIdx.x;
  if (idx >= (long long)n * 64) return;
  int c = (int)(idx & 63);
  float invN = 1.0f / (float)n;
  float mean = stats[c] * invN;
  float var  = stats[64 + c] * invN - mean * mean;
  float v = (h[idx] + bias[c] - mean) * rsqrtf(var + 1e-5f) * gamma[c] + beta[c];
  h[idx] = fmaxf(v, 0.0f);
}

// ------------------------------------------------------- graph mean pooling
__global__ void k_pool(const float* __restrict__ h, const float* __restrict__ b3,
                       const int* __restrict__ batch, float* __restrict__ psum,
                       int* __restrict__ pcnt, int n) {
  long long idx = (long long)blockIdx.x * 256 + threadIdx.x;
  if (idx >= (long long)n * 64) return;
  int node = (int)(idx >> 6), c = (int)(idx & 63);
  int g = batch[node];
  atomic_add_f32(&psum[(size_t)g * 64 + c], h[idx] + b3[c]);
  if (c == 0) atomicAdd(&pcnt[g], 1);
}

// --------------------------------------------------- head MLP, 1 block/graph
__global__ __launch_bounds__(128) void k_head(const float* __restrict__ psum,
                                              const int* __restrict__ pcnt,
                                              const float* __restrict__ gf,
                                              const float* __restrict__ Wg,  const float* __restrict__ bg,
                                              const float* __restrict__ Wp1, const float* __restrict__ bp1,
                                              const float* __restrict__ Wp2, const float* __restrict__ bp2,
                                              float* __restrict__ out) {
  __shared__ float comb[96];
  __shared__ float o1[64];
  int g = blockIdx.x, t = threadIdx.x;
  if (t < 64) {
    float cnt = (float)pcnt[g];
    comb[t] = psum[(size_t)g * 64 + t] / fmaxf(cnt, 1.0f);
  } else if (t < 96) {
    int j = t - 64;
    float acc = bg[j];
    for (int k = 0; k < 10; ++k) acc += gf[g * 10 + k] * Wg[k * 32 + j];
    comb[t] = fmaxf(acc, 0.0f);
  }
  __syncthreads();
  if (t < 64) {
    float acc = bp1[t];
    for (int k = 0; k < 96; ++k) acc += comb[k] * Wp1[k * 64 + t];
    o1[t] = fmaxf(acc, 0.0f);
  }
  __syncthreads();
  if (t < 5) {
    float acc = bp2[t];
    for (int k = 0; k < 64; ++k) acc += o1[k] * Wp2[k * 5 + t];
    out[g * 5 + t] = acc;
  }
}

// ---------------------------------------------------------------- launcher
extern "C" void kernel_launch(void* const* d_in, const int* in_sizes, int n_in,
                              void* d_out, int out_size, void* d_ws, size_t ws_size,
                              hipStream_t stream) {
  const float* x   = (const float*)d_in[0];
  const float* gf  = (const float*)d_in[1];
  const float* W1  = (const float*)d_in[2];
  const float* b1  = (const float*)d_in[3];
  const float* W2  = (const float*)d_in[4];
  const float* b2  = (const float*)d_in[5];
  const float* W3  = (const float*)d_in[6];
  const float* b3  = (const float*)d_in[7];
  const float* g1  = (const float*)d_in[8];
  const float* be1 = (const float*)d_in[9];
  const float* g2  = (const float*)d_in[10];
  const float* be2 = (const float*)d_in[11];
  const float* Wg  = (const float*)d_in[12];
  const float* bg  = (const float*)d_in[13];
  const float* Wp1 = (const float*)d_in[14];
  const float* bp1 = (const float*)d_in[15];
  const float* Wp2 = (const float*)d_in[16];
  const float* bp2 = (const float*)d_in[17];
  const int* ei    = (const int*)d_in[18];  // [2,E] int32 (JAX x64 disabled)
  const int* batch = (const int*)d_in[19];

  const int N = in_sizes[0] / 7;
  const int E = in_sizes[18] / 2;
  const int G = in_sizes[1] / 10;
  const int* src = ei;
  const int* dst = ei + E;

  // workspace carve (256B aligned): ~52 MB
  char* w = (char*)d_ws;
  auto alloc = [&](size_t bytes) -> char* {
    char* p = w; w += (bytes + 255) & ~(size_t)255; return p;
  };
  float* hA    = (float*)alloc((size_t)N * 64 * 4);  // features / agg (in-place)
  float* hB    = (float*)alloc((size_t)N * 64 * 4);  // GEMM output
  float* dis   = (float*)alloc((size_t)N * 4);
  int*   deg   = (int*)  alloc((size_t)N * 4);
  float* stats = (float*)alloc(128 * 4);             // [sum(64) | sumsq(64)]
  float* psum  = (float*)alloc((size_t)G * 64 * 4);
  int*   pcnt  = (int*)  alloc((size_t)G * 4);

  auto cdiv = [](long long a, long long b) { return (int)((a + b - 1) / b); };
  const dim3 B(256);
  const long long NH = (long long)N * 64;

  // degrees -> dis
  k_zero<<<cdiv(N, 256), B, 0, stream>>>((float*)deg, N);
  k_count_deg<<<cdiv(E, 256), B, 0, stream>>>(dst, deg, E);
  k_dis<<<cdiv(N, 256), B, 0, stream>>>(deg, dis, N);

  // ---- layer 1 ----
  k_linear_in<<<cdiv(NH, 256), B, 0, stream>>>(x, W1, hB, N);
  k_init_agg<<<cdiv(NH, 256), B, 0, stream>>>(hB, dis, hA, N);
  k_scatter<<<cdiv((long long)E * 16, 256), B, 0, stream>>>(hB, src, dst, dis, hA, E);
  k_zero<<<1, 128, 0, stream>>>(stats, 128);
  k_stats<<<cdiv(N, 512), B, 0, stream>>>(hA, b1, stats, N);
  k_bn_relu<<<cdiv(NH, 256), B, 0, stream>>>(hA, b1, stats, g1, be1, N);

  // ---- layer 2 ----
  k_linear_wmma<<<cdiv(N, 128), B, 0, stream>>>(hA, W2, hB, N);
  k_init_agg<<<cdiv(NH, 256), B, 0, stream>>>(hB, dis, hA, N);
  k_scatter<<<cdiv((long long)E * 16, 256), B, 0, stream>>>(hB, src, dst, dis, hA, E);
  k_zero<<<1, 128, 0, stream>>>(stats, 128);
  k_stats<<<cdiv(N, 512), B, 0, stream>>>(hA, b2, stats, N);
  k_bn_relu<<<cdiv(NH, 256), B, 0, stream>>>(hA, b2, stats, g2, be2, N);

  // ---- layer 3 (no BN) ----
  k_linear_wmma<<<cdiv(N, 128), B, 0, stream>>>(hA, W3, hB, N);
  k_init_agg<<<cdiv(NH, 256), B, 0, stream>>>(hB, dis, hA, N);
  k_scatter<<<cdiv((long long)E * 16, 256), B, 0, stream>>>(hB, src, dst, dis, hA, E);

  // ---- pool + head ----
  k_zero<<<cdiv(G * 64, 256), B, 0, stream>>>(psum, G * 64);
  k_zero<<<cdiv(G, 256), B, 0, stream>>>((float*)pcnt, G);
  k_pool<<<cdiv(NH, 256), B, 0, stream>>>(hA, b3, batch, psum, pcnt, N);
  k_head<<<G, 128, 0, stream>>>(psum, pcnt, gf, Wg, bg, Wp1, bp1, Wp2, bp2, (float*)d_out);
}